// Seq2Seq_16578573763422
// MI455X (gfx1250) — compile-verified
//
#include <hip/hip_runtime.h>

// ---------------------------------------------------------------------------
// Seq2Seq LSTM (enc H=4 x2, dec H=10 x2) for MI455X / gfx1250.
// One wave32 owns 16 batch columns; gate GEMMs run on V_WMMA_F32_16X16X4_F32
// (fp32 in/out, matching the fp32 reference). State lives in registers,
// replicated across the two 16-lane halves so WMMA B operands are built with
// simple selects and gate gather needs only shfl_xor(16).
// Round-1 fixes: native v_tanh_f32 / v_rcp_f32 activations (no IEEE divide
// chains), software-pipelined x prefetch in the encoder loop.
// ---------------------------------------------------------------------------

typedef __attribute__((ext_vector_type(2))) float v2f;
typedef __attribute__((ext_vector_type(8))) float v8f;

#define WMMA4(A, Bv, C) \
  __builtin_amdgcn_wmma_f32_16x16x4_f32(false, (A), false, (Bv), (short)0, (C), false, false)

static constexpr int BN = 4096;
static constexpr int TN = 512;

// ---- fast activations: gfx1250 has V_TANH_F32; avoid precise-divide chains.
#if __has_builtin(__builtin_amdgcn_tanhf)
__device__ __forceinline__ float ftanh(float x) { return __builtin_amdgcn_tanhf(x); }
#else
__device__ __forceinline__ float ftanh(float x) {
  // tanh(x) = 1 - 2/(exp(2x)+1), with a single v_rcp_f32
  return 1.0f - 2.0f * __builtin_amdgcn_rcpf(__expf(2.0f * x) + 1.0f);
}
#endif
__device__ __forceinline__ float fsig(float x) {
  // sigmoid(x) = 0.5*tanh(x/2) + 0.5  (1 trans op when v_tanh exists)
  return fmaf(0.5f, ftanh(0.5f * x), 0.5f);
}

// B-operand: lane supplies features {2*half, 2*half+1} of a K=4 chunk.
__device__ __forceinline__ v2f bsel(int half, float f0, float f1, float f2, float f3) {
  v2f r;
  r.x = half ? f2 : f0;
  r.y = half ? f3 : f1;
  return r;
}

// A-operand element: combined [Wih | Whh | bias | 0...] feature matrix, row g.
__device__ __forceinline__ float wfeat(const float* Wih, const float* Whh,
                                       const float* bih, const float* bhh,
                                       int in_dim, int H, int g, int j) {
  if (g >= 4 * H) return 0.0f;
  if (j < in_dim) return Wih[g * in_dim + j];
  j -= in_dim;
  if (j < H) return Whh[g * H + j];
  j -= H;
  return (j == 0) ? (bih[g] + bhh[g]) : 0.0f;
}

__device__ __forceinline__ float xswap(float v) { return __shfl_xor(v, 16, 32); }

// Gate row r (0..15) of a single 16x16 tile, for this lane's batch column.
__device__ __forceinline__ float row16(const v8f& a, const v8f& p, int half, int r) {
  const int v = r & 7, rh = r >> 3;
  return (rh == half) ? a[v] : p[v];
}
// Gate row r (0..47) across three stacked 16-row tiles.
__device__ __forceinline__ float row48(const v8f* a, const v8f* p, int half, int r) {
  const int tt = r >> 4, m = r & 15, v = m & 7, rh = m >> 3;
  return (rh == half) ? a[tt][v] : p[tt][v];
}

__global__ __launch_bounds__(32) void seq2seq_lstm_wmma(
    const float* __restrict__ x,
    const float* __restrict__ eW0i, const float* __restrict__ eW0h,
    const float* __restrict__ eb0i, const float* __restrict__ eb0h,
    const float* __restrict__ eW1i, const float* __restrict__ eW1h,
    const float* __restrict__ eb1i, const float* __restrict__ eb1h,
    const float* __restrict__ dW0i, const float* __restrict__ dW0h,
    const float* __restrict__ db0i, const float* __restrict__ db0h,
    const float* __restrict__ dW1i, const float* __restrict__ dW1h,
    const float* __restrict__ db1i, const float* __restrict__ db1h,
    const float* __restrict__ fcW, const float* __restrict__ fcb,
    const float* __restrict__ outW, const float* __restrict__ outb,
    float* __restrict__ outFinal,   // (B,T,2)
    float* __restrict__ outFwd,     // (B,2)
    float* __restrict__ outNoise)   // (B,T,1)
{
  const int lane = threadIdx.x & 31;
  const int half = lane >> 4;
  const int rr   = lane & 15;
  const int b    = blockIdx.x * 16 + rr;

  // ---- preload WMMA A operands (weight tiles, bias folded as feature) ----
  v2f aE0[2], aE1[3], aD0[3][3], aD1[3][6];
#pragma unroll
  for (int c = 0; c < 2; ++c) {
    aE0[c].x = wfeat(eW0i, eW0h, eb0i, eb0h, 1, 4, rr, 4 * c + 2 * half);
    aE0[c].y = wfeat(eW0i, eW0h, eb0i, eb0h, 1, 4, rr, 4 * c + 2 * half + 1);
  }
#pragma unroll
  for (int c = 0; c < 3; ++c) {
    aE1[c].x = wfeat(eW1i, eW1h, eb1i, eb1h, 4, 4, rr, 4 * c + 2 * half);
    aE1[c].y = wfeat(eW1i, eW1h, eb1i, eb1h, 4, 4, rr, 4 * c + 2 * half + 1);
  }
#pragma unroll
  for (int tt = 0; tt < 3; ++tt)
#pragma unroll
    for (int c = 0; c < 3; ++c) {
      aD0[tt][c].x = wfeat(dW0i, dW0h, db0i, db0h, 1, 10, 16 * tt + rr, 4 * c + 2 * half);
      aD0[tt][c].y = wfeat(dW0i, dW0h, db0i, db0h, 1, 10, 16 * tt + rr, 4 * c + 2 * half + 1);
    }
#pragma unroll
  for (int tt = 0; tt < 3; ++tt)
#pragma unroll
    for (int c = 0; c < 6; ++c) {
      aD1[tt][c].x = wfeat(dW1i, dW1h, db1i, db1h, 10, 10, 16 * tt + rr, 4 * c + 2 * half);
      aD1[tt][c].y = wfeat(dW1i, dW1h, db1i, db1h, 10, 10, 16 * tt + rr, 4 * c + 2 * half + 1);
    }

  const v8f vzero = {0.f, 0.f, 0.f, 0.f, 0.f, 0.f, 0.f, 0.f};

  // ---- encoder: state replicated in both lane halves ----
  float h0[4] = {0.f, 0.f, 0.f, 0.f}, c0[4] = {0.f, 0.f, 0.f, 0.f};
  float h1[4] = {0.f, 0.f, 0.f, 0.f}, c1[4] = {0.f, 0.f, 0.f, 0.f};

  const float* xrow = x + (size_t)b * TN;
  float xt = xrow[0];  // software pipeline: x[t] loaded one iteration ahead

  for (int t = 0; t < TN; ++t) {
    const float xcur = xt;
    if (t + 1 < TN) xt = xrow[t + 1];  // issue next load; waited next iter

    // layer 0: features [x, h0[0..3], 1, 0, 0]
    v8f acc = vzero;
    acc = WMMA4(aE0[0], bsel(half, xcur, h0[0], h0[1], h0[2]), acc);
    acc = WMMA4(aE0[1], bsel(half, h0[3], 1.f, 0.f, 0.f), acc);
    v8f pac;
#pragma unroll
    for (int v = 0; v < 8; ++v) pac[v] = xswap(acc[v]);
#pragma unroll
    for (int k = 0; k < 4; ++k) {
      const float gi = row16(acc, pac, half, k);
      const float gf = row16(acc, pac, half, 4 + k);
      const float gg = row16(acc, pac, half, 8 + k);
      const float go = row16(acc, pac, half, 12 + k);
      c0[k] = fsig(gf) * c0[k] + fsig(gi) * ftanh(gg);
      h0[k] = fsig(go) * ftanh(c0[k]);
    }

    // layer 1: features [h0new[0..3], h1[0..3], 1, 0, 0, 0]
    acc = vzero;
    acc = WMMA4(aE1[0], bsel(half, h0[0], h0[1], h0[2], h0[3]), acc);
    acc = WMMA4(aE1[1], bsel(half, h1[0], h1[1], h1[2], h1[3]), acc);
    acc = WMMA4(aE1[2], bsel(half, 1.f, 0.f, 0.f, 0.f), acc);
#pragma unroll
    for (int v = 0; v < 8; ++v) pac[v] = xswap(acc[v]);
#pragma unroll
    for (int k = 0; k < 4; ++k) {
      const float gi = row16(acc, pac, half, k);
      const float gf = row16(acc, pac, half, 4 + k);
      const float gg = row16(acc, pac, half, 8 + k);
      const float go = row16(acc, pac, half, 12 + k);
      c1[k] = fsig(gf) * c1[k] + fsig(gi) * ftanh(gg);
      h1[k] = fsig(go) * ftanh(c1[k]);
    }
  }

  // ---- fwd_out = h1 @ fc_W.T + fc_b ----
  float fwd0 = fcb[0], fwd1 = fcb[1];
#pragma unroll
  for (int k = 0; k < 4; ++k) {
    fwd0 = fmaf(fcW[k], h1[k], fwd0);
    fwd1 = fmaf(fcW[4 + k], h1[k], fwd1);
  }
  if (!half) {
    outFwd[(size_t)b * 2 + 0] = fwd0;
    outFwd[(size_t)b * 2 + 1] = fwd1;
  }

  // ---- decoder: seed with zero-padded encoder state ----
  float hd0[10], cd0[10], hd1[10], cd1[10];
#pragma unroll
  for (int k = 0; k < 10; ++k) {
    hd0[k] = (k < 4) ? h0[k] : 0.f;
    cd0[k] = (k < 4) ? c0[k] : 0.f;
    hd1[k] = (k < 4) ? h1[k] : 0.f;
    cd1[k] = (k < 4) ? c1[k] : 0.f;
  }

  const float ow0 = outW[0], ow1 = outW[1], ow2 = outW[2], ow3 = outW[3], ow4 = outW[4];
  const float ow5 = outW[5], ow6 = outW[6], ow7 = outW[7], ow8 = outW[8], ow9 = outW[9];
  const float ob = outb[0];

  for (int t = 0; t < TN; ++t) {
    const float inp = hd1[0];

    // decoder layer 0: 40 gates in 3 M-tiles; features [inp, hd0[0..9], 1]
    v8f d[3];
    d[0] = vzero; d[1] = vzero; d[2] = vzero;
    {
      v2f bch = bsel(half, inp, hd0[0], hd0[1], hd0[2]);
#pragma unroll
      for (int tt = 0; tt < 3; ++tt) d[tt] = WMMA4(aD0[tt][0], bch, d[tt]);
      bch = bsel(half, hd0[3], hd0[4], hd0[5], hd0[6]);
#pragma unroll
      for (int tt = 0; tt < 3; ++tt) d[tt] = WMMA4(aD0[tt][1], bch, d[tt]);
      bch = bsel(half, hd0[7], hd0[8], hd0[9], 1.f);
#pragma unroll
      for (int tt = 0; tt < 3; ++tt) d[tt] = WMMA4(aD0[tt][2], bch, d[tt]);
    }
    v8f pd[3];
#pragma unroll
    for (int tt = 0; tt < 3; ++tt)
#pragma unroll
      for (int v = 0; v < 8; ++v) pd[tt][v] = xswap(d[tt][v]);
#pragma unroll
    for (int k = 0; k < 10; ++k) {
      const float gi = row48(d, pd, half, k);
      const float gf = row48(d, pd, half, 10 + k);
      const float gg = row48(d, pd, half, 20 + k);
      const float go = row48(d, pd, half, 30 + k);
      cd0[k] = fsig(gf) * cd0[k] + fsig(gi) * ftanh(gg);
      hd0[k] = fsig(go) * ftanh(cd0[k]);
    }

    // decoder layer 1: features [hd0new[0..9], hd1[0..9], 1, 0, 0, 0]
    d[0] = vzero; d[1] = vzero; d[2] = vzero;
    {
      v2f bch = bsel(half, hd0[0], hd0[1], hd0[2], hd0[3]);
#pragma unroll
      for (int tt = 0; tt < 3; ++tt) d[tt] = WMMA4(aD1[tt][0], bch, d[tt]);
      bch = bsel(half, hd0[4], hd0[5], hd0[6], hd0[7]);
#pragma unroll
      for (int tt = 0; tt < 3; ++tt) d[tt] = WMMA4(aD1[tt][1], bch, d[tt]);
      bch = bsel(half, hd0[8], hd0[9], hd1[0], hd1[1]);
#pragma unroll
      for (int tt = 0; tt < 3; ++tt) d[tt] = WMMA4(aD1[tt][2], bch, d[tt]);
      bch = bsel(half, hd1[2], hd1[3], hd1[4], hd1[5]);
#pragma unroll
      for (int tt = 0; tt < 3; ++tt) d[tt] = WMMA4(aD1[tt][3], bch, d[tt]);
      bch = bsel(half, hd1[6], hd1[7], hd1[8], hd1[9]);
#pragma unroll
      for (int tt = 0; tt < 3; ++tt) d[tt] = WMMA4(aD1[tt][4], bch, d[tt]);
      bch = bsel(half, 1.f, 0.f, 0.f, 0.f);
#pragma unroll
      for (int tt = 0; tt < 3; ++tt) d[tt] = WMMA4(aD1[tt][5], bch, d[tt]);
    }
#pragma unroll
    for (int tt = 0; tt < 3; ++tt)
#pragma unroll
      for (int v = 0; v < 8; ++v) pd[tt][v] = xswap(d[tt][v]);
#pragma unroll
    for (int k = 0; k < 10; ++k) {
      const float gi = row48(d, pd, half, k);
      const float gf = row48(d, pd, half, 10 + k);
      const float gg = row48(d, pd, half, 20 + k);
      const float go = row48(d, pd, half, 30 + k);
      cd1[k] = fsig(gf) * cd1[k] + fsig(gi) * ftanh(gg);
      hd1[k] = fsig(go) * ftanh(cd1[k]);
    }

    // noise = hd1 @ out_W.T + out_b ; final = noise + fwd_out broadcast
    float nv = ob;
    nv = fmaf(ow0, hd1[0], nv); nv = fmaf(ow1, hd1[1], nv);
    nv = fmaf(ow2, hd1[2], nv); nv = fmaf(ow3, hd1[3], nv);
    nv = fmaf(ow4, hd1[4], nv); nv = fmaf(ow5, hd1[5], nv);
    nv = fmaf(ow6, hd1[6], nv); nv = fmaf(ow7, hd1[7], nv);
    nv = fmaf(ow8, hd1[8], nv); nv = fmaf(ow9, hd1[9], nv);

    if (!half) {
      const size_t bt = (size_t)b * TN + t;
      outNoise[bt] = nv;
      outFinal[bt * 2 + 0] = nv + fwd0;
      outFinal[bt * 2 + 1] = nv + fwd1;
    }
  }
}

extern "C" void kernel_launch(void* const* d_in, const int* in_sizes, int n_in,
                              void* d_out, int out_size, void* d_ws, size_t ws_size,
                              hipStream_t stream) {
  (void)in_sizes; (void)n_in; (void)out_size; (void)d_ws; (void)ws_size;

  const float* x    = (const float*)d_in[0];
  // d_in[1] = context (unused by reference)
  const float* eW0i = (const float*)d_in[2];
  const float* eW0h = (const float*)d_in[3];
  const float* eb0i = (const float*)d_in[4];
  const float* eb0h = (const float*)d_in[5];
  const float* eW1i = (const float*)d_in[6];
  const float* eW1h = (const float*)d_in[7];
  const float* eb1i = (const float*)d_in[8];
  const float* eb1h = (const float*)d_in[9];
  const float* dW0i = (const float*)d_in[10];
  const float* dW0h = (const float*)d_in[11];
  const float* db0i = (const float*)d_in[12];
  const float* db0h = (const float*)d_in[13];
  const float* dW1i = (const float*)d_in[14];
  const float* dW1h = (const float*)d_in[15];
  const float* db1i = (const float*)d_in[16];
  const float* db1h = (const float*)d_in[17];
  const float* fcW  = (const float*)d_in[18];
  const float* fcb  = (const float*)d_in[19];
  const float* outW = (const float*)d_in[20];
  const float* outb = (const float*)d_in[21];

  float* out      = (float*)d_out;
  float* outFinal = out;                                          // B*T*2
  float* outFwd   = out + (size_t)BN * TN * 2;                    // B*2
  float* outNoise = out + (size_t)BN * TN * 2 + (size_t)BN * 2;   // B*T

  dim3 grid(BN / 16), block(32);
  seq2seq_lstm_wmma<<<grid, block, 0, stream>>>(
      x, eW0i, eW0h, eb0i, eb0h, eW1i, eW1h, eb1i, eb1h,
      dW0i, dW0h, db0i, db0h, dW1i, dW1h, db1i, db1h,
      fcW, fcb, outW, outb, outFinal, outFwd, outNoise);
}